// GaborRendererCUDA_76287209112274
// MI455X (gfx1250) — compile-verified
//
#include <hip/hip_runtime.h>

#define CHUNK 256          // atoms staged per LDS chunk (== blockDim.x)
#define BLOCK 256
#define SPT 4              // samples per thread
#define WAVE_SPAN 128      // 32 lanes * SPT contiguous samples per wave
#define TILE (BLOCK * SPT) // 1024 samples per block
#define ATOM_GROUPS 8

__launch_bounds__(BLOCK)
__global__ void gabor_render_kernel(const float* __restrict__ g_amp,
                                    const float* __restrict__ g_tau,
                                    const float* __restrict__ g_omega,
                                    const float* __restrict__ g_sigma,
                                    const float* __restrict__ g_phi,
                                    const float* __restrict__ g_gamma,
                                    float* __restrict__ out,
                                    int N, int T) {
    __shared__ float raw[6][CHUNK];   // async-staged raw parameters
    __shared__ float drv[8][CHUNK];   // derived per-atom constants (SoA)

    const int tid  = threadIdx.x;
    const int lane = tid & 31;
    // wave id made explicitly wave-uniform (scalar) for coherent skip branches
    const int w = __builtin_amdgcn_readfirstlane(tid >> 5);

    const int b0    = blockIdx.x * TILE;
    const int wbase = b0 + w * WAVE_SPAN;

    const float inv_sr = 1.0f / 24000.0f;
    float ts[SPT];
#pragma unroll
    for (int k = 0; k < SPT; ++k)
        ts[k] = (float)(wbase + lane + 32 * k) * inv_sr;

    // wave time-interval center/half-width for the uniform skip test
    const float wc = ((float)wbase + (float)(WAVE_SPAN - 1) * 0.5f) * inv_sr;
    const float wh = ((float)WAVE_SPAN * 0.5f) * inv_sr;

    float acc[SPT];
#pragma unroll
    for (int k = 0; k < SPT; ++k) acc[k] = 0.0f;

    const float* params[6] = {g_amp, g_tau, g_omega, g_sigma, g_phi, g_gamma};

    const int nChunks = (N + CHUNK - 1) / CHUNK;
    for (int c = blockIdx.y; c < nChunks; c += gridDim.y) {
        __syncthreads();  // previous iteration finished reading raw/drv

        // ---- async global -> LDS staging of the 6 parameter arrays ----
        const int gidx    = c * CHUNK + tid;
        const int clamped = (gidx < N) ? gidx : (N - 1);
        const unsigned goff = (unsigned)clamped * 4u;
#pragma unroll
        for (int p = 0; p < 6; ++p) {
            // low 32 bits of a generic LDS pointer == wave-relative LDS byte offset
            unsigned ldsa = (unsigned)(unsigned long long)(&raw[p][tid]);
            asm volatile("global_load_async_to_lds_b32 %0, %1, %2"
                         :
                         : "v"(ldsa), "v"(goff), "s"(params[p])
                         : "memory");
        }
        asm volatile("s_wait_asynccnt 0" ::: "memory");
        __syncthreads();

        // ---- derived per-atom constants (one thread per atom) ----
        {
            float a  = raw[0][tid];
            float tu = raw[1][tid];
            float om = raw[2][tid];
            float sg = raw[3][tid];
            float ph = raw[4][tid];
            float gm = raw[5][tid];
            bool valid = (gidx < N);
            drv[0][tid] = tu;                                        // tau
            drv[1][tid] = 4.0f * sg;                                 // 4*sigma (window)
            drv[2][tid] = valid ? a : 0.0f;                          // amplitude
            drv[3][tid] = om;                                        // omega (Hz = rev/s)
            drv[4][tid] = 0.5f * gm;                                 // gamma/2 (rev/s^2)
            drv[5][tid] = -1.44269504f / (2.0f * sg * sg + 1e-8f);   // -log2e/(2s^2+eps)
            drv[6][tid] = ph * 0.15915494f;                          // phi / (2*pi)
            drv[7][tid] = 16.0f * sg * sg;                           // (4*sigma)^2 mask
        }
        __syncthreads();

        // ---- hot loop over staged atoms ----
        for (int a = 0; a < CHUNK; ++a) {
            float atau = drv[0][a];
            float s4   = drv[1][a];
            // wave-coherent window test: all lanes agree -> execz skip
            if (__builtin_fabsf(wc - atau) > s4 + wh) continue;

            float aamp = drv[2][a];
            float om   = drv[3][a];
            float hg   = drv[4][a];
            float ce   = drv[5][a];
            float pr   = drv[6][a];
            float r2   = drv[7][a];
#pragma unroll
            for (int k = 0; k < SPT; ++k) {
                float tc  = ts[k] - atau;
                float tc2 = tc * tc;
                float e   = __builtin_amdgcn_exp2f(tc2 * ce);          // v_exp_f32
                // phase in revolutions: omega*tc + 0.5*gamma*tc^2 + phi/2pi
                float rev = __builtin_fmaf(__builtin_fmaf(hg, tc, om), tc, pr);
                rev -= __builtin_rintf(rev);                           // v_rndne range-reduce
                float cs  = __builtin_amdgcn_cosf(rev);                // v_cos_f32 (turns)
                float v   = aamp * e * cs;
                acc[k] += (tc2 <= r2) ? v : 0.0f;
            }
        }
    }

#pragma unroll
    for (int k = 0; k < SPT; ++k) {
        int sidx = wbase + lane + 32 * k;
        if (sidx < T) atomicAdd(&out[sidx], acc[k]);   // global_atomic_add_f32
    }
}

extern "C" void kernel_launch(void* const* d_in, const int* in_sizes, int n_in,
                              void* d_out, int out_size, void* d_ws, size_t ws_size,
                              hipStream_t stream) {
    const float* amp   = (const float*)d_in[0];
    const float* tau   = (const float*)d_in[1];
    const float* omega = (const float*)d_in[2];
    const float* sigma = (const float*)d_in[3];
    const float* phi   = (const float*)d_in[4];
    const float* gamma = (const float*)d_in[5];
    const int N = in_sizes[0];
    const int T = out_size;   // num_samples == out_size

    hipMemsetAsync(d_out, 0, (size_t)T * sizeof(float), stream);

    dim3 grid((T + TILE - 1) / TILE, ATOM_GROUPS);
    gabor_render_kernel<<<grid, BLOCK, 0, stream>>>(amp, tau, omega, sigma, phi, gamma,
                                                    (float*)d_out, N, T);
}